// AttentionPooling3_6717328851082
// MI455X (gfx1250) — compile-verified
//
#include <hip/hip_runtime.h>
#include <math.h>
#include <stdint.h>

#define BATCH   16
#define NHEAD   8
#define SEQ     4096
#define IDIM    1024
#define VDIM    128
#define KDIM    64
#define CH      8                  // T-chunks per (b,h)
#define TC      (SEQ / CH)         // 512 rows per block
#define NWAVES  8
#define NTHREADS 256
#define NROWS   (TC / NWAVES)      // 64 rows per wave
#define DEPTH   8                  // async pipeline depth (rows in flight per wave)
#define PART_STRIDE (VDIM + 2)     // [m, l, acc[128]]
#define LOG2E_F 1.4426950408889634f

// ---- CDNA5 async-copy path (guarded probe) --------------------------------
#if defined(__has_builtin)
#  if __has_builtin(__builtin_amdgcn_global_load_async_to_lds_b128) && \
      __has_builtin(__builtin_amdgcn_s_wait_asynccnt)
#    define HAVE_ASYNC 1
#  endif
#endif
#ifndef HAVE_ASYNC
#  define HAVE_ASYNC 0
#endif

#if defined(__has_builtin)
#  if __has_builtin(__builtin_amdgcn_exp2f)
#    define EXP2F __builtin_amdgcn_exp2f
#  endif
#endif
#ifndef EXP2F
#  define EXP2F exp2f
#endif

#if HAVE_ASYNC
// Builtin signature (from hipcc diagnostic): param0 is int4 AS(1)*.
typedef int v4i_cdna5 __attribute__((ext_vector_type(4)));
typedef __attribute__((address_space(1))) v4i_cdna5* g_v4i_ptr;
typedef __attribute__((address_space(3))) v4i_cdna5* l_v4i_ptr;

__device__ __forceinline__ void async_ld16(const float* g, float4* lds) {
  // global_load_async_to_lds_b128: 16 B per lane, tracked by ASYNCcnt
  __builtin_amdgcn_global_load_async_to_lds_b128(
      (g_v4i_ptr)(g), (l_v4i_ptr)(lds), /*offset=*/0, /*cpol=*/0);
}
#define WAIT_ASYNC(n) __builtin_amdgcn_s_wait_asynccnt(n)
#endif

// ---- Pass 1: per-(b,h,chunk) online-softmax partial -----------------------
extern "C" __global__ __launch_bounds__(NTHREADS) void
attnpool_partial(const float* __restrict__ x, const float* __restrict__ Q,
                 const float* __restrict__ keyp, float* __restrict__ part)
{
  const int h    = blockIdx.x;          // 0..7
  const int b    = blockIdx.y;          // 0..15
  const int c    = blockIdx.z;          // 0..CH-1
  const int tid  = threadIdx.x;
  const int wid  = tid >> 5;
  const int lane = tid & 31;

  __shared__ __align__(16) float  s_w[VDIM];
#if HAVE_ASYNC
  __shared__ __align__(16) float4 s_buf[NWAVES * DEPTH * 32];   // 32 KB
#endif
  __shared__ float s_m[NWAVES];
  __shared__ float s_l[NWAVES];
  __shared__ __align__(16) float s_acc[NWAVES][VDIM];

  // Row base for this (b, h, chunk); wave wid handles t = base + wid + 8*i
  const float* xbase =
      x + (size_t)(b * SEQ + c * TC) * IDIM + h * VDIM + lane * 4;

#if HAVE_ASYNC
  // Prime the async pipeline FIRST: these loads don't depend on w, so the DMA
  // overlaps the w computation + barrier below (hides startup HBM latency).
  float4* mybuf = &s_buf[wid * (DEPTH * 32)];
  auto issue = [&](int i) {
    const float* g = xbase + (size_t)(wid + NWAVES * i) * IDIM;
    async_ld16(g, mybuf + (i & (DEPTH - 1)) * 32 + lane);
  };
#pragma unroll
  for (int i = 0; i < DEPTH; ++i) issue(i);
#endif

  // w_h[v] = sum_k Q[h,v,k] * key_p[h,k]   (collapses the whole projection)
  if (tid < VDIM) {
    const float* q = Q + ((size_t)h * VDIM + tid) * KDIM;
    const float* p = keyp + (size_t)h * KDIM;
    float s = 0.f;
#pragma unroll 16
    for (int k = 0; k < KDIM; ++k) s = fmaf(q[k], p[k], s);
    s_w[tid] = s;
  }
  __syncthreads();
  const float4 w4 = *(const float4*)(&s_w[lane * 4]);   // lane owns v = 4*lane..4*lane+3

  float  m   = -__builtin_huge_valf();
  float  l   = 0.f;
  float4 acc = make_float4(0.f, 0.f, 0.f, 0.f);

  auto step = [&](float4 xv) {
    // per-lane partial dot, then wave32 xor-butterfly allreduce
    float d = fmaf(xv.x, w4.x, fmaf(xv.y, w4.y, fmaf(xv.z, w4.z, xv.w * w4.w)));
    d += __shfl_xor(d, 16);
    d += __shfl_xor(d, 8);
    d += __shfl_xor(d, 4);
    d += __shfl_xor(d, 2);
    d += __shfl_xor(d, 1);
    const float logit = d * 0.125f;                 // 1/sqrt(KDIM)
    if (logit <= m) {                               // common case: no rescale
      const float e = EXP2F((logit - m) * LOG2E_F);
      l += e;
      acc.x = fmaf(e, xv.x, acc.x);
      acc.y = fmaf(e, xv.y, acc.y);
      acc.z = fmaf(e, xv.z, acc.z);
      acc.w = fmaf(e, xv.w, acc.w);
    } else {                                        // rare: new max, rescale
      const float s = EXP2F((m - logit) * LOG2E_F); // exp2(-inf)=0 handles init
      l = fmaf(l, s, 1.f);
      acc.x = fmaf(acc.x, s, xv.x);
      acc.y = fmaf(acc.y, s, xv.y);
      acc.z = fmaf(acc.z, s, xv.z);
      acc.w = fmaf(acc.w, s, xv.w);
      m = logit;
    }
  };

#if HAVE_ASYNC
  for (int i = 0; i < NROWS - DEPTH; ++i) {
    WAIT_ASYNC(DEPTH - 1);                          // oldest slot complete
    asm volatile("" ::: "memory");
    const float4 xv = mybuf[(i & (DEPTH - 1)) * 32 + lane];
    asm volatile("" ::: "memory");                  // keep LDS read before refill
    issue(i + DEPTH);                               // refill same slot
    step(xv);
  }
  WAIT_ASYNC(0);                                    // drain tail
  asm volatile("" ::: "memory");
  for (int i = NROWS - DEPTH; i < NROWS; ++i) {
    const float4 xv = mybuf[(i & (DEPTH - 1)) * 32 + lane];
    step(xv);
  }
#else
  for (int i = 0; i < NROWS; ++i) {
    const float4 xv = *(const float4*)(xbase + (size_t)(wid + NWAVES * i) * IDIM);
    step(xv);
  }
#endif

  // combine the 8 wave-local softmax states within the block
  *(float4*)(&s_acc[wid][lane * 4]) = acc;
  if (lane == 0) { s_m[wid] = m; s_l[wid] = l; }
  __syncthreads();

  if (tid < VDIM) {
    float M = s_m[0];
#pragma unroll
    for (int w = 1; w < NWAVES; ++w) M = fmaxf(M, s_m[w]);
    float L = 0.f, a = 0.f;
#pragma unroll
    for (int w = 0; w < NWAVES; ++w) {
      const float sc = EXP2F((s_m[w] - M) * LOG2E_F);
      L = fmaf(s_l[w], sc, L);
      a = fmaf(s_acc[w][tid], sc, a);
    }
    float* dst = part + (size_t)((b * NHEAD + h) * CH + c) * PART_STRIDE;
    dst[2 + tid] = a;
    if (tid == 0) dst[0] = M;
    if (tid == 1) dst[1] = L;
  }
}

// ---- Pass 2: merge CH chunk partials per (b,h), write output --------------
extern "C" __global__ __launch_bounds__(VDIM) void
attnpool_final(const float* __restrict__ part, float* __restrict__ out)
{
  const int h = blockIdx.x;
  const int b = blockIdx.y;
  const int v = threadIdx.x;
  const float* base = part + (size_t)((b * NHEAD + h) * CH) * PART_STRIDE;

  float M = base[0];
#pragma unroll
  for (int c = 1; c < CH; ++c)
    M = fmaxf(M, base[(size_t)c * PART_STRIDE]);

  float L = 0.f, a = 0.f;
#pragma unroll
  for (int c = 0; c < CH; ++c) {
    const float* pc = base + (size_t)c * PART_STRIDE;
    const float sc = EXP2F((pc[0] - M) * LOG2E_F);
    L = fmaf(pc[1], sc, L);
    a = fmaf(pc[2 + v], sc, a);
  }
  out[(size_t)b * IDIM + h * VDIM + v] = a / L;
}

// ---- Host launcher --------------------------------------------------------
extern "C" void kernel_launch(void* const* d_in, const int* in_sizes, int n_in,
                              void* d_out, int out_size, void* d_ws, size_t ws_size,
                              hipStream_t stream) {
  const float* x  = (const float*)d_in[0];   // [16, 4096, 1024] f32
  const float* Q  = (const float*)d_in[1];   // [8, 128, 64] f32
  const float* kp = (const float*)d_in[2];   // [8, 64, 1] f32
  float* out  = (float*)d_out;               // [16, 1024] f32
  float* part = (float*)d_ws;                // needs 16*8*8*130*4 = 520 KiB

  dim3 g1(NHEAD, BATCH, CH);
  attnpool_partial<<<g1, NTHREADS, 0, stream>>>(x, Q, kp, part);

  dim3 g2(NHEAD, BATCH);
  attnpool_final<<<g2, VDIM, 0, stream>>>(part, out);

  (void)in_sizes; (void)n_in; (void)out_size; (void)ws_size;
}